// GINNet_49727131353730
// MI455X (gfx1250) — compile-verified
//
#include <hip/hip_runtime.h>
#include <hip/hip_bf16.h>

#define H 128
#define NGRAPHS 100
#define NCLASS 10
#define BN_EPS 1e-5f

typedef float v2f  __attribute__((ext_vector_type(2)));
typedef float v8f  __attribute__((ext_vector_type(8)));

// ---------------------------------------------------------------------------
// Utility: zero a float buffer
// ---------------------------------------------------------------------------
__global__ void k_zero(float* __restrict__ p, int n) {
    int i = blockIdx.x * blockDim.x + threadIdx.x;
    if (i < n) p[i] = 0.0f;
}

// ---------------------------------------------------------------------------
// GEMM + fused column stats:
//   out[N x 128] = A[N x 128] @ W[128 x 128] + bias[128]
//   optional: osum[f] += sum_r out[r][f], osq[f] += sum_r out[r][f]^2
// Block = 256 threads = 8 waves, covering 32 rows x 128 cols.
// Wave w owns cols [16w,16w+16) with TWO 16x16 accumulators (rows row0..+15
// and row0+16..+31) sharing one B fragment per K-step -> 2 chained-independent
// v_wmma_f32_16x16x4_f32 per step, 32 steps over K=128.
// A-frag (16x4 f32): lanes0-15 M=lane (v0=K0,v1=K1); lanes16-31 (v0=K2,v1=K3).
// C/D: lane&15 = column; vgpr r -> row r (lanes0-15) / r+8 (lanes16-31).
// ---------------------------------------------------------------------------
__global__ __launch_bounds__(256) void k_gemm(const float* __restrict__ A,
                                              const float* __restrict__ W,
                                              const float* __restrict__ bias,
                                              float* __restrict__ out,
                                              float* __restrict__ osum,
                                              float* __restrict__ osq,
                                              int N) {
    __shared__ float red[512];
    const int tid  = threadIdx.x;
    const int wave = tid >> 5;
    const int lane = tid & 31;
    const int m    = lane & 15;
    const int hi   = lane >> 4;
    const int kh   = hi * 2;            // K-pair base: 0 or 2
    const int row0 = blockIdx.x * 32;
    const int col0 = wave * 16;

    int r0 = row0 + m;      if (r0 >= N) r0 = N - 1;
    int r1 = row0 + 16 + m; if (r1 >= N) r1 = N - 1;
    const float* __restrict__ A0 = A + (size_t)r0 * H;
    const float* __restrict__ A1 = A + (size_t)r1 * H;
    const float* __restrict__ Wc = W + col0 + m;

    v8f acc0 = {}, acc1 = {};
    #pragma unroll 4
    for (int k = 0; k < H; k += 4) {
        v2f a0, a1, b;
        float2 t0 = *(const float2*)(A0 + k + kh);
        float2 t1 = *(const float2*)(A1 + k + kh);
        a0.x = t0.x; a0.y = t0.y;
        a1.x = t1.x; a1.y = t1.y;
        b.x = Wc[(size_t)(k + kh)     * H];
        b.y = Wc[(size_t)(k + kh + 1) * H];
        acc0 = __builtin_amdgcn_wmma_f32_16x16x4_f32(
            false, a0, false, b, (short)0, acc0, false, false);
        acc1 = __builtin_amdgcn_wmma_f32_16x16x4_f32(
            false, a1, false, b, (short)0, acc1, false, false);
    }

    const float bv  = bias[col0 + m];
    const int   rb0 = row0 + hi * 8;        // rows for acc0
    const int   rb1 = row0 + 16 + hi * 8;   // rows for acc1
    float s = 0.0f, q = 0.0f;

    if (row0 + 32 <= N) {                   // fast path: no guards
        #pragma unroll
        for (int r = 0; r < 8; ++r) {
            float v0 = acc0[r] + bv;
            float v1 = acc1[r] + bv;
            out[(size_t)(rb0 + r) * H + col0 + m] = v0;
            out[(size_t)(rb1 + r) * H + col0 + m] = v1;
            s += v0 + v1;
            q += v0 * v0 + v1 * v1;
        }
    } else {                                // single partial block
        #pragma unroll
        for (int r = 0; r < 8; ++r) {
            float v0 = acc0[r] + bv;
            float v1 = acc1[r] + bv;
            if (rb0 + r < N) { out[(size_t)(rb0 + r) * H + col0 + m] = v0; s += v0; q += v0 * v0; }
            if (rb1 + r < N) { out[(size_t)(rb1 + r) * H + col0 + m] = v1; s += v1; q += v1 * v1; }
        }
    }

    if (osum) {
        red[tid]       = s;
        red[256 + tid] = q;
        __syncthreads();
        if (lane < 16) {   // one thread per column: combine lane m & m+16
            atomicAdd(osum + col0 + m, red[tid] + red[tid + 16]);
            atomicAdd(osq  + col0 + m, red[256 + tid] + red[256 + tid + 16]);
        }
    }
}

// ---------------------------------------------------------------------------
// z = (1 + eps[layer]) * h     (vectorized float4)
// ---------------------------------------------------------------------------
__global__ void k_scale(const float* __restrict__ h, float* __restrict__ z,
                        const float* __restrict__ eps, int layer, int n4) {
    int i = blockIdx.x * blockDim.x + threadIdx.x;
    if (i >= n4) return;
    float s = 1.0f + eps[layer];
    float4 v = ((const float4*)h)[i];
    v.x *= s; v.y *= s; v.z *= s; v.w *= s;
    ((float4*)z)[i] = v;
}

// ---------------------------------------------------------------------------
// Scatter-add: z[dst[e]] += h[src[e]]   (one wave per edge, float4 gather,
// per-float global f32 atomics -> global_atomic_add_f32)
// ---------------------------------------------------------------------------
__global__ void k_scatter(const float* __restrict__ h,
                          const int* __restrict__ src,
                          const int* __restrict__ dst,
                          float* __restrict__ z, int E) {
    int idx  = blockIdx.x * blockDim.x + threadIdx.x;
    int e    = idx >> 5;
    int lane = idx & 31;
    if (e >= E) return;
    int s = src[e];
    int d = dst[e];
    float4 v = ((const float4*)(h + (size_t)s * H))[lane];
    float* zd = z + (size_t)d * H + lane * 4;
    atomicAdd(zd + 0, v.x);
    atomicAdd(zd + 1, v.y);
    atomicAdd(zd + 2, v.z);
    atomicAdd(zd + 3, v.w);
}

// ---------------------------------------------------------------------------
// Fused BN epilogue. Block covers 64 rows; thread = (feature f, row-phase ro).
//   v  = relu(gamma[f]*(x-mu)*rsqrt(var+eps)+beta[f])      (mu,var from isum/isq)
//   if snorm :  v *= snorm[row]
//   if resid :  pv = resid += v   (x untouched)   else x = v, pv = v
//   if osum  :  accumulate column sum/sumsq of v (LDS reduce, 1 atomic/feat/blk)
//   if pooled:  pooled[gids[row]][f] += pv  (run-length compressed: gids sorted)
// ---------------------------------------------------------------------------
__global__ __launch_bounds__(256) void k_bn_fuse(
    float* __restrict__ x,
    const float* __restrict__ isum, const float* __restrict__ isq,
    const float* __restrict__ gamma, const float* __restrict__ beta,
    const float* __restrict__ snorm,   // nullable
    float* __restrict__ resid,         // nullable
    float* __restrict__ osum,          // nullable (with osq)
    float* __restrict__ osq,
    const int* __restrict__ gids,      // nullable (with pooled)
    float* __restrict__ pooled,
    int N, float invN) {
    __shared__ float red[512];
    const int f    = threadIdx.x & (H - 1);
    const int ro   = threadIdx.x >> 7;
    const int row0 = blockIdx.x * 64;
    const int rend = (row0 + 64 < N) ? row0 + 64 : N;

    const float mu  = isum[f] * invN;
    const float var = isq[f] * invN - mu * mu;
    const float sc  = gamma[f] * rsqrtf(var + BN_EPS);
    const float sh  = beta[f] - mu * sc;      // v = sc*x + sh

    float s = 0.0f, q = 0.0f;
    int   cur_g = -1;
    float pacc  = 0.0f;

    for (int r = row0 + ro; r < rend; r += 2) {
        size_t idx = (size_t)r * H + f;
        float v = fmaxf(sc * x[idx] + sh, 0.0f);
        if (snorm) v *= snorm[r];
        float pv = v;
        if (resid) { pv = resid[idx] + v; resid[idx] = pv; }
        else       { x[idx] = v; }
        s += v;
        q += v * v;
        if (pooled) {
            int g = gids[r];
            if (g != cur_g) {
                if (cur_g >= 0) atomicAdd(&pooled[(size_t)cur_g * H + f], pacc);
                cur_g = g;
                pacc  = 0.0f;
            }
            pacc += pv;
        }
    }
    if (pooled && cur_g >= 0) atomicAdd(&pooled[(size_t)cur_g * H + f], pacc);

    if (osum) {
        red[threadIdx.x]       = s;
        red[256 + threadIdx.x] = q;
        __syncthreads();
        if (ro == 0) {
            atomicAdd(&osum[f], red[f] + red[f + H]);
            atomicAdd(&osq[f],  red[256 + f] + red[256 + f + H]);
        }
    }
}

// ---------------------------------------------------------------------------
// Graph pooling for hidden_rep[0]: pooled[gid[n]][f] += h[n][f]
// (run-length compressed atomics; graph_ids are sorted)
// ---------------------------------------------------------------------------
__global__ __launch_bounds__(256) void k_pool(const float* __restrict__ h,
                                              const int* __restrict__ gids,
                                              float* __restrict__ pooled, int N) {
    const int f    = threadIdx.x & (H - 1);
    const int ro   = threadIdx.x >> 7;
    const int row0 = blockIdx.x * 64;
    const int rend = (row0 + 64 < N) ? row0 + 64 : N;
    int   cur_g = -1;
    float acc   = 0.0f;
    for (int r = row0 + ro; r < rend; r += 2) {
        int g = gids[r];
        if (g != cur_g) {
            if (cur_g >= 0) atomicAdd(&pooled[(size_t)cur_g * H + f], acc);
            cur_g = g;
            acc   = 0.0f;
        }
        acc += h[(size_t)r * H + f];
    }
    if (cur_g >= 0) atomicAdd(&pooled[(size_t)cur_g * H + f], acc);
}

// ---------------------------------------------------------------------------
// score[100 x 10] += pooled[100 x 128] @ Wp[128 x 10] + bp[10]
// ---------------------------------------------------------------------------
__global__ void k_pred(const float* __restrict__ pooled,
                       const float* __restrict__ Wp,
                       const float* __restrict__ bp,
                       float* __restrict__ score) {
    int idx = blockIdx.x * blockDim.x + threadIdx.x;
    if (idx >= NGRAPHS * NCLASS) return;
    int g = idx / NCLASS;
    int c = idx % NCLASS;
    float s = bp[c];
    #pragma unroll 8
    for (int k = 0; k < H; ++k)
        s += pooled[(size_t)g * H + k] * Wp[(size_t)k * NCLASS + c];
    score[idx] += s;
}

// ---------------------------------------------------------------------------
extern "C" void kernel_launch(void* const* d_in, const int* in_sizes, int n_in,
                              void* d_out, int out_size, void* d_ws, size_t ws_size,
                              hipStream_t stream) {
    const float* h_in  = (const float*)d_in[0];
    const float* snorm = (const float*)d_in[1];
    const int*   esrc  = (const int*)d_in[2];
    const int*   edst  = (const int*)d_in[3];
    const int*   gids  = (const int*)d_in[4];
    const float* embW  = (const float*)d_in[5];
    const float* embB  = (const float*)d_in[6];
    const float* eps   = (const float*)d_in[7];
    const float* W1    = (const float*)d_in[8];
    const float* b1    = (const float*)d_in[9];
    const float* g1    = (const float*)d_in[10];
    const float* be1   = (const float*)d_in[11];
    const float* W2    = (const float*)d_in[12];
    const float* b2    = (const float*)d_in[13];
    const float* ga    = (const float*)d_in[14];
    const float* ba    = (const float*)d_in[15];
    const float* gl    = (const float*)d_in[16];
    const float* bl    = (const float*)d_in[17];
    const float* predW = (const float*)d_in[18];
    const float* predB = (const float*)d_in[19];
    float* score = (float*)d_out;

    const int N = in_sizes[1];   // snorm_n has N elements
    const int E = in_sizes[2];
    const int L = 4;

    // Workspace layout (floats)
    float* ws     = (float*)d_ws;
    size_t off    = 0;
    float* hbuf   = ws + off; off += (size_t)N * H;
    float* zbuf   = ws + off; off += (size_t)N * H;
    float* tbuf   = ws + off; off += (size_t)N * H;
    float* pooled = ws + off; off += (size_t)NGRAPHS * H;
    float* statsA = ws + off; off += 2 * H;   // [sum | sumsq]
    float* statsB = ws + off; off += 2 * H;

    const int   nElem = N * H;
    const int   n4    = nElem / 4;
    const dim3  blk(256);
    const int   gGemm = (N + 31) / 32;
    const int   gRows = (N + 63) / 64;
    const int   gVec4 = (n4 + 255) / 256;
    const int   gEdge = (E * 32 + 255) / 256;
    const int   gPred = (NGRAPHS * NCLASS + 255) / 256;
    const float invN  = 1.0f / (float)N;

    // score = 0
    k_zero<<<(NGRAPHS * NCLASS + 255) / 256, blk, 0, stream>>>(score, NGRAPHS * NCLASS);

    // embedding_h: hbuf = h_in @ embW + embB
    k_gemm<<<gGemm, blk, 0, stream>>>(h_in, embW, embB, hbuf, nullptr, nullptr, N);

    // hidden_rep[0] pooling + prediction
    k_zero<<<(NGRAPHS * H + 255) / 256, blk, 0, stream>>>(pooled, NGRAPHS * H);
    k_pool<<<gRows, blk, 0, stream>>>(hbuf, gids, pooled, N);
    k_pred<<<gPred, blk, 0, stream>>>(pooled, predW, predB, score);

    for (int l = 0; l < L; ++l) {
        const float* W1l = W1 + (size_t)l * H * H;
        const float* W2l = W2 + (size_t)l * H * H;

        // z = (1+eps)*h ; z += scatter(h)
        k_scale<<<gVec4, blk, 0, stream>>>(hbuf, zbuf, eps, l, n4);
        k_scatter<<<gEdge, blk, 0, stream>>>(hbuf, esrc, edst, zbuf, E);

        // t = z @ W1 + b1 (fused stats) ; BN(g1,be1)+relu in-place
        k_zero<<<1, blk, 0, stream>>>(statsA, 2 * H);
        k_gemm<<<gGemm, blk, 0, stream>>>(zbuf, W1l, b1 + l * H, tbuf,
                                          statsA, statsA + H, N);
        k_bn_fuse<<<gRows, blk, 0, stream>>>(tbuf, statsA, statsA + H,
                                             g1 + l * H, be1 + l * H,
                                             nullptr, nullptr, nullptr, nullptr,
                                             nullptr, nullptr, N, invN);

        // z = t @ W2 + b2 (fused stats) ; BN(ga,ba)+relu ; *= snorm (fused
        // output stats -> statsB for next BN)
        k_zero<<<1, blk, 0, stream>>>(statsA, 2 * H);
        k_gemm<<<gGemm, blk, 0, stream>>>(tbuf, W2l, b2 + l * H, zbuf,
                                          statsA, statsA + H, N);
        k_zero<<<1, blk, 0, stream>>>(statsB, 2 * H);
        k_bn_fuse<<<gRows, blk, 0, stream>>>(zbuf, statsA, statsA + H,
                                             ga + l * H, ba + l * H,
                                             snorm, nullptr, statsB, statsB + H,
                                             nullptr, nullptr, N, invN);

        // BN(gl,bl)+relu ; residual h += v ; fused pooling of h_new
        k_zero<<<(NGRAPHS * H + 255) / 256, blk, 0, stream>>>(pooled, NGRAPHS * H);
        k_bn_fuse<<<gRows, blk, 0, stream>>>(zbuf, statsB, statsB + H,
                                             gl + l * H, bl + l * H,
                                             nullptr, hbuf, nullptr, nullptr,
                                             gids, pooled, N, invN);

        // prediction for hidden_rep[l+1]
        k_pred<<<gPred, blk, 0, stream>>>(pooled,
                                          predW + (size_t)(l + 1) * H * NCLASS,
                                          predB + (size_t)(l + 1) * NCLASS, score);
    }
}